// Attention_39238821216394
// MI455X (gfx1250) — compile-verified
//
#include <hip/hip_runtime.h>

typedef float v2f __attribute__((ext_vector_type(2)));
typedef float v8f __attribute__((ext_vector_type(8)));

#define BATCH 32
#define NHEAD 32
#define KVH   8
#define GQA   4        // NHEAD / KVH
#define HDIM  128
#define BS    256      // tokens per physical block
#define BPS   8        // blocks per sequence
#define SCALE 0.08838834764831845f
#define NEG_BIG (-1e30f)

__device__ __forceinline__ v8f wmma4(v2f a, v2f b, v8f c) {
  // D = A(16x4,f32) * B(4x16,f32) + C(16x16,f32)
  return __builtin_amdgcn_wmma_f32_16x16x4_f32(false, a, false, b, (short)0, c,
                                               false, false);
}

__device__ __forceinline__ float rmax16(float x) {
#pragma unroll
  for (int off = 8; off >= 1; off >>= 1) x = fmaxf(x, __shfl_xor(x, off, 16));
  return x;
}
__device__ __forceinline__ float rsum16(float x) {
#pragma unroll
  for (int off = 8; off >= 1; off >>= 1) x += __shfl_xor(x, off, 16);
  return x;
}

__global__ __launch_bounds__(256) void paged_attn_wmma(
    const float* __restrict__ q, const float* __restrict__ knew_g,
    const float* __restrict__ vnew_g, const float* __restrict__ kcache,
    const float* __restrict__ vcache, const int* __restrict__ block_tables,
    const int* __restrict__ context_lens, float* __restrict__ out) {
  __shared__ __align__(16) float Qs[16][HDIM];   // 8 KB, rows 4..15 = 0
  __shared__ __align__(16) float Ps[8][16][16];  // 8 KB, per-wave P tile
  __shared__ float Os[8][GQA][HDIM];             // 16 KB, per-wave partial O
  __shared__ float Ml[8][GQA][2];                // per-wave (m, l)

  const int g = blockIdx.x;          // kv head
  const int b = blockIdx.y;          // batch
  const int tid = threadIdx.x;
  const int w = tid >> 5;            // wave id 0..7 == block-in-seq
  const int lane = tid & 31;
  const int ln = lane & 15;
  const bool hi = lane >= 16;
  const int koff = hi ? 2 : 0;

  // ---- stage Q (4 real rows, 12 zero rows) into LDS ----
  for (int i = tid; i < 16 * HDIM; i += 256) {
    int r = i >> 7, d = i & 127;
    Qs[r][d] = (r < GQA) ? q[((size_t)b * NHEAD + g * GQA + r) * HDIM + d]
                         : 0.0f;
  }
  __syncthreads();

  const int ctx = context_lens[b];
  const int last = ctx - 1;
  const int wave_base = w * BS;
  const int phys = block_tables[b * BPS + w];
  const float* kb = kcache + ((size_t)phys * BS * KVH + g) * HDIM;
  const float* vb = vcache + ((size_t)phys * BS * KVH + g) * HDIM;
  const float* knew = knew_g + ((size_t)b * KVH + g) * HDIM;
  const float* vnew = vnew_g + ((size_t)b * KVH + g) * HDIM;

  float m_run[GQA], l_run[GQA];
  v8f acc[8];
#pragma unroll
  for (int r = 0; r < GQA; ++r) { m_run[r] = NEG_BIG; l_run[r] = 0.0f; }
#pragma unroll
  for (int nb = 0; nb < 8; ++nb)
#pragma unroll
    for (int i = 0; i < 8; ++i) acc[nb][i] = 0.0f;

  for (int tile = 0; tile < 16; ++tile) {           // 16 tokens per tile
    const int tb = tile * 16;
    if (wave_base + tb >= ctx) break;               // wave-uniform exit

    // ---- scores: S = Q * K^T, accumulate K-dim in steps of 4 ----
    const int toff = tb + ln;                       // this lane's token column
    const int tglob = wave_base + toff;
    const float* krow =
        (tglob == last) ? knew : (kb + (size_t)toff * (KVH * HDIM));
    v8f s;
#pragma unroll
    for (int i = 0; i < 8; ++i) s[i] = 0.0f;
#pragma unroll 8
    for (int kk = 0; kk < 32; ++kk) {
      v2f a = *(const v2f*)(&Qs[ln][4 * kk + koff]);       // A: Q[ln][k..k+1]
      v2f bf = *(const v2f*)(krow + 4 * kk + koff);        // B: K[tok][k..k+1]
      s = wmma4(a, bf, s);
    }

    // ---- online softmax (rows 0..3 real; 4..15 harmless garbage) ----
    const bool valid = tglob < ctx;
    float alpha[GQA];
#pragma unroll
    for (int r = 0; r < GQA; ++r) {
      float raw = s[r] * SCALE;
      float val = valid ? raw : NEG_BIG;
      float mt = rmax16(val);
      float mnew = fmaxf(m_run[r], mt);
      float a8 = __expf(m_run[r] - mnew);                  // -1e30 -> 0, no NaN
      float p = valid ? __expf(raw - mnew) : 0.0f;
      l_run[r] = l_run[r] * a8 + rsum16(p);
      m_run[r] = mnew;
      alpha[r] = a8;
      s[r] = p;
    }
#pragma unroll
    for (int r = 4; r < 8; ++r)                            // zero-Q rows
      s[r] = valid ? __expf(s[r] * SCALE - m_run[r - 4]) : 0.0f;

    // ---- P tile -> per-wave LDS (re-fragment as A operand) ----
#pragma unroll
    for (int i = 0; i < 8; ++i) Ps[w][hi ? (i + 8) : i][ln] = s[i];

    // ---- rescale running O ----
#pragma unroll
    for (int nb = 0; nb < 8; ++nb)
#pragma unroll
      for (int i = 0; i < 8; ++i) acc[nb][i] *= alpha[i & 3];

    // ---- O += P * V : 4 k-steps x 8 dim-tiles ----
#pragma unroll
    for (int j = 0; j < 4; ++j) {
      v2f a = *(const v2f*)(&Ps[w][ln][4 * j + koff]);     // A: P[ln][tok..+1]
      const int o0 = tb + 4 * j + koff;
      const int g0 = wave_base + o0, g1 = g0 + 1;
      const float* r0 = (g0 == last) ? vnew : vb + (size_t)o0 * (KVH * HDIM);
      const float* r1 = (g1 == last) ? vnew : vb + (size_t)(o0 + 1) * (KVH * HDIM);
#pragma unroll
      for (int nb = 0; nb < 8; ++nb) {
        v2f bf;
        bf.x = r0[nb * 16 + ln];                           // B: V[tok][dim]
        bf.y = r1[nb * 16 + ln];
        acc[nb] = wmma4(a, bf, acc[nb]);
      }
    }
  }

  // ---- publish per-wave partials ----
#pragma unroll
  for (int nb = 0; nb < 8; ++nb)
    if (!hi)
#pragma unroll
      for (int i = 0; i < GQA; ++i) Os[w][i][nb * 16 + ln] = acc[nb][i];
  if (lane == 0)
#pragma unroll
    for (int r = 0; r < GQA; ++r) { Ml[w][r][0] = m_run[r]; Ml[w][r][1] = l_run[r]; }
  __syncthreads();

  // ---- cross-wave log-sum-exp combine, write out[b, g*4+r, :] ----
  for (int idx = tid; idx < GQA * HDIM; idx += 256) {
    int r = idx >> 7, d = idx & 127;
    float M = NEG_BIG;
#pragma unroll
    for (int ww = 0; ww < 8; ++ww) M = fmaxf(M, Ml[ww][r][0]);
    float L = 0.0f, val = 0.0f;
#pragma unroll
    for (int ww = 0; ww < 8; ++ww) {
      float we = __expf(Ml[ww][r][0] - M);
      L += we * Ml[ww][r][1];
      val += we * Os[ww][r][d];
    }
    out[((size_t)b * NHEAD + g * GQA + r) * HDIM + d] = val / L;
  }
}

extern "C" void kernel_launch(void* const* d_in, const int* in_sizes, int n_in,
                              void* d_out, int out_size, void* d_ws,
                              size_t ws_size, hipStream_t stream) {
  (void)in_sizes; (void)n_in; (void)d_ws; (void)ws_size; (void)out_size;
  const float* q = (const float*)d_in[0];
  const float* k = (const float*)d_in[1];
  const float* v = (const float*)d_in[2];
  const float* kcache = (const float*)d_in[3];
  const float* vcache = (const float*)d_in[4];
  // d_in[5] = slot_mapping (unused: the (ctx-1) token is redirected to k/v)
  const int* block_tables = (const int*)d_in[6];
  const int* context_lens = (const int*)d_in[7];
  float* out = (float*)d_out;

  dim3 grid(KVH, BATCH);  // 256 workgroups, one per (kv-head, batch)
  paged_attn_wmma<<<grid, 256, 0, stream>>>(q, k, v, kcache, vcache,
                                            block_tables, context_lens, out);
}